// FasterRCNNRegress_63101659513040
// MI455X (gfx1250) — compile-verified
//
#include <hip/hip_runtime.h>
#include <math.h>

// ---------------- problem constants ----------------
#define KB 4          // images
#define KP 2000       // proposals/image
#define KNC 6000      // candidates/image = P*(C-1)
#define KNS 6016      // candidates padded to 32*188 (mask column count)
#define KNSORT 8192   // sort size (pow2 >= 6000)
#define KMW 188       // mask words per row (6016/32)
#define KMS 192       // mask row stride in words (async-load friendly)
#define KTOPK 100
#define IMGW 800.0f
#define IMGH 800.0f
#define SCORE_T 0.05f
#define NMS_T 0.5f
#define MINSZ 0.01f
#define BHC 4.1351665567423560f   // log(1000/16)

typedef float v2f __attribute__((ext_vector_type(2)));
typedef float v8f __attribute__((ext_vector_type(8)));

// ---------------- workspace layout (bytes) ----------------
static constexpr size_t O_MAXC   = 0;                                   // KB u32
static constexpr size_t O_CBOX   = 256;                                 // KB*KNC float4
static constexpr size_t O_CSCORE = O_CBOX   + (size_t)KB*KNC*16;
static constexpr size_t O_CLABEL = O_CSCORE + (size_t)KB*KNC*4;
static constexpr size_t O_CLEN   = O_CLABEL + (size_t)KB*KNC*4;
static constexpr size_t O_CVALID = O_CLEN   + (size_t)KB*KNC*4;
static constexpr size_t O_ORDER  = O_CVALID + (size_t)KB*KNC*4;         // KB*KNSORT int
static constexpr size_t O_SOFF   = O_ORDER  + (size_t)KB*KNSORT*4;      // KB*KNS float4 (offset boxes)
static constexpr size_t O_SBOX   = O_SOFF   + (size_t)KB*KNS*16;        // KB*KNS float4 (clipped boxes)
static constexpr size_t O_SSCORE = O_SBOX   + (size_t)KB*KNS*16;
static constexpr size_t O_SLABEL = O_SSCORE + (size_t)KB*KNS*4;
static constexpr size_t O_SLEN   = O_SLABEL + (size_t)KB*KNS*4;
static constexpr size_t O_SVALID = O_SLEN   + (size_t)KB*KNS*4;
static constexpr size_t O_SAREA  = O_SVALID + (size_t)KB*KNS*4;
static constexpr size_t O_MASK   = ((O_SAREA + (size_t)KB*KNS*4 + 255) / 256) * 256;
// mask: KB * KNC rows * KMS words (+4KB tail pad read-over by async b128)

// ---------------- K0: init per-image max coordinate ----------------
__global__ void k_init(char* __restrict__ ws) {
  if (threadIdx.x < KB) ((unsigned*)(ws + O_MAXC))[threadIdx.x] = 0u;
}

// ---------------- K1: decode + softmax + clip + validity ----------------
__global__ void k_decode(const float* __restrict__ cls, const float* __restrict__ reg,
                         const float* __restrict__ lenl, const float* __restrict__ prop,
                         char* __restrict__ ws) {
  int t = blockIdx.x * blockDim.x + threadIdx.x;
  if (t >= KB * KP) return;
  int b = t / KP, p = t % KP;

  const float* lg = cls + (size_t)t * 4;
  float l0 = lg[0], l1 = lg[1], l2 = lg[2], l3 = lg[3];
  float mx = fmaxf(fmaxf(l0, l1), fmaxf(l2, l3));
  float e0 = expf(l0 - mx), e1 = expf(l1 - mx), e2 = expf(l2 - mx), e3 = expf(l3 - mx);
  float inv = 1.0f / (e0 + e1 + e2 + e3);
  float sc[4] = {e0 * inv, e1 * inv, e2 * inv, e3 * inv};

  const float* pr = prop + (size_t)t * 4;
  float w  = pr[2] - pr[0], h = pr[3] - pr[1];
  float cx = pr[0] + 0.5f * w, cy = pr[1] + 0.5f * h;
  float length = lenl[t];
  const float* rg = reg + (size_t)t * 16;

  float4*   cbox   = (float4*)(ws + O_CBOX);
  float*    cscore = (float*)(ws + O_CSCORE);
  int*      clabel = (int*)(ws + O_CLABEL);
  float*    clen   = (float*)(ws + O_CLEN);
  int*      cvalid = (int*)(ws + O_CVALID);
  unsigned* maxc   = (unsigned*)(ws + O_MAXC);

  #pragma unroll
  for (int c = 1; c < 4; ++c) {
    float dx = rg[4 * c + 0] / 10.0f;
    float dy = rg[4 * c + 1] / 10.0f;
    float dw = fminf(rg[4 * c + 2] / 5.0f, BHC);
    float dh = fminf(rg[4 * c + 3] / 5.0f, BHC);
    float pcx = dx * w + cx, pcy = dy * h + cy;
    float pw = expf(dw) * w, ph = expf(dh) * h;
    float x1 = fminf(fmaxf(pcx - 0.5f * pw, 0.0f), IMGW);
    float y1 = fminf(fmaxf(pcy - 0.5f * ph, 0.0f), IMGH);
    float x2 = fminf(fmaxf(pcx + 0.5f * pw, 0.0f), IMGW);
    float y2 = fminf(fmaxf(pcy + 0.5f * ph, 0.0f), IMGH);
    int vd = (sc[c] > SCORE_T) && ((x2 - x1) >= MINSZ) && ((y2 - y1) >= MINSZ);
    int ci = b * KNC + p * 3 + (c - 1);
    cbox[ci]   = make_float4(x1, y1, x2, y2);
    cscore[ci] = sc[c];
    clabel[ci] = c;
    clen[ci]   = length;
    cvalid[ci] = vd;
    if (vd) {  // clipped coords are >= 0 so uint-bit atomicMax == float max
      atomicMax(maxc + b, __float_as_uint(x1));
      atomicMax(maxc + b, __float_as_uint(y1));
      atomicMax(maxc + b, __float_as_uint(x2));
      atomicMax(maxc + b, __float_as_uint(y2));
    }
  }
}

// ---------------- K2: per-image bitonic argsort (score desc, idx asc) ----------------
__global__ void k_sort(char* __restrict__ ws) {
  extern __shared__ unsigned long long skey[];   // KNSORT u64
  int b = blockIdx.x;
  const float* cs = (const float*)(ws + O_CSCORE) + (size_t)b * KNC;
  const int*   cv = (const int*)(ws + O_CVALID) + (size_t)b * KNC;
  for (int i = threadIdx.x; i < KNSORT; i += blockDim.x) {
    float key = __uint_as_float(0xFF800000u);            // -inf
    if (i < KNC && cv[i]) key = cs[i];
    unsigned ub = __float_as_uint(key);
    unsigned u  = (ub & 0x80000000u) ? ~ub : (ub | 0x80000000u);  // order-preserving
    skey[i] = ((unsigned long long)(~u) << 32) | (unsigned)i;     // asc u64 == desc score, asc idx
  }
  __syncthreads();
  for (unsigned k = 2; k <= KNSORT; k <<= 1) {
    for (unsigned j = k >> 1; j > 0; j >>= 1) {
      for (unsigned i = threadIdx.x; i < KNSORT; i += blockDim.x) {
        unsigned ixj = i ^ j;
        if (ixj > i) {
          unsigned long long a = skey[i], c = skey[ixj];
          bool up = ((i & k) == 0);
          if ((a > c) == up) { skey[i] = c; skey[ixj] = a; }
        }
      }
      __syncthreads();
    }
  }
  int* order = (int*)(ws + O_ORDER) + (size_t)b * KNSORT;
  for (int i = threadIdx.x; i < KNSORT; i += blockDim.x)
    order[i] = (int)(unsigned)(skey[i] & 0xFFFFFFFFull);
}

// ---------------- K3: gather sorted arrays + batched-NMS offset boxes + areas ----------------
__global__ void k_gather(char* __restrict__ ws) {
  int t = blockIdx.x * blockDim.x + threadIdx.x;
  if (t >= KB * KNS) return;
  int b = t / KNS, pos = t % KNS;
  float4* soff = (float4*)(ws + O_SOFF) + (size_t)b * KNS;
  float4* sbox = (float4*)(ws + O_SBOX) + (size_t)b * KNS;
  float*  ssc  = (float*)(ws + O_SSCORE) + (size_t)b * KNS;
  int*    slb  = (int*)(ws + O_SLABEL) + (size_t)b * KNS;
  float*  sln  = (float*)(ws + O_SLEN) + (size_t)b * KNS;
  int*    svd  = (int*)(ws + O_SVALID) + (size_t)b * KNS;
  float*  sar  = (float*)(ws + O_SAREA) + (size_t)b * KNS;
  if (pos < KNC) {
    int o = ((const int*)(ws + O_ORDER))[(size_t)b * KNSORT + pos];
    float4 bx = ((const float4*)(ws + O_CBOX))[(size_t)b * KNC + o];
    float sc = ((const float*)(ws + O_CSCORE))[(size_t)b * KNC + o];
    int   lb = ((const int*)(ws + O_CLABEL))[(size_t)b * KNC + o];
    float ln = ((const float*)(ws + O_CLEN))[(size_t)b * KNC + o];
    int   vd = ((const int*)(ws + O_CVALID))[(size_t)b * KNC + o];
    float mc = __uint_as_float(((const unsigned*)(ws + O_MAXC))[b]);
    float off = (float)lb * (mc + 1.0f);
    float ox1 = bx.x + off, oy1 = bx.y + off, ox2 = bx.z + off, oy2 = bx.w + off;
    soff[pos] = make_float4(ox1, oy1, ox2, oy2);
    sbox[pos] = bx;
    ssc[pos] = sc; slb[pos] = lb; sln[pos] = ln; svd[pos] = vd;
    sar[pos] = (ox2 - ox1) * (oy2 - oy1);
  } else {
    soff[pos] = make_float4(0, 0, 0, 0);
    sbox[pos] = make_float4(0, 0, 0, 0);
    ssc[pos] = 0; slb[pos] = 0; sln[pos] = 0; svd[pos] = 0; sar[pos] = 0;
  }
}

// ---------------- K4: suppression bitmask, one wave per 16x32 tile ----------------
// union additive term (area_i + area_j) for each 16x16 tile computed on the
// matrix unit: D = A x B with A=[area_row,1,0,0] (16x4), B=[1;area_col;0;0] (4x16)
// -> v_wmma_f32_16x16x4_f32 accumulator lands in exactly the C layout each lane
// consumes for its 8 (row,col) pairs.
__global__ void k_mask(char* __restrict__ ws) {
  int wave = (blockIdx.x * blockDim.x + threadIdx.x) >> 5;
  int lane = threadIdx.x & 31;
  const int ROWT = KNC / 16;                 // 375
  const int TILES = KB * ROWT * KMW;
  if (wave >= TILES) return;
  int ct = wave % KMW;
  int rt = (wave / KMW) % ROWT;
  int b  = wave / (KMW * ROWT);
  int rowBase = rt * 16, colBase = ct * 32;
  unsigned* mrow = (unsigned*)(ws + O_MASK) + (size_t)b * KNC * KMS;

  if (colBase + 31 <= rowBase) {             // strictly lower triangle: all-zero words
    if (lane < 16) mrow[(size_t)(rowBase + lane) * KMS + ct] = 0u;
    return;
  }

  int half = lane >> 4, ln = lane & 15;
  const float4* sob = (const float4*)(ws + O_SOFF) + (size_t)b * KNS;
  const float*  sar = (const float*)(ws + O_SAREA) + (size_t)b * KNS;
  int colA = colBase + ln, colB = colBase + 16 + ln;
  float4 cbA = sob[colA], cbB = sob[colB];
  float  aA = sar[colA], aB = sar[colB];
  float  rowArea = (half == 0) ? sar[rowBase + ln] : 0.0f;

  // A (16x4 f32): lanes0-15 VGPR0=K0 VGPR1=K1 ; lanes16-31 K2,K3 (zero)
  v2f A;  A.x  = rowArea;                 A.y  = (half == 0) ? 1.0f : 0.0f;
  v2f B0; B0.x = (half == 0) ? 1.0f : 0; B0.y = (half == 0) ? aA : 0.0f;
  v2f B1; B1.x = B0.x;                   B1.y = (half == 0) ? aB : 0.0f;
  v8f z = {};
  v8f acc0 = __builtin_amdgcn_wmma_f32_16x16x4_f32(false, A, false, B0, (short)0, z, false, false);
  v8f acc1 = __builtin_amdgcn_wmma_f32_16x16x4_f32(false, A, false, B1, (short)0, z, false, false);

  unsigned bitsA = 0, bitsB = 0;
  #pragma unroll
  for (int v = 0; v < 8; ++v) {
    int rowIdx = rowBase + v + 8 * half;   // matches C-layout M for this lane/VGPR
    float4 rb = sob[rowIdx];
    {
      float iw = fminf(rb.z, cbA.z) - fmaxf(rb.x, cbA.x);
      float ih = fminf(rb.w, cbA.w) - fmaxf(rb.y, cbA.y);
      float inter = fmaxf(iw, 0.0f) * fmaxf(ih, 0.0f);
      float uni = acc0[v] - inter;         // area_i + area_j - inter (WMMA outer-sum)
      bool sup = (uni > 0.0f) && (inter > NMS_T * uni) && (colA > rowIdx) && (colA < KNC);
      bitsA |= (sup ? 1u : 0u) << v;
    }
    {
      float iw = fminf(rb.z, cbB.z) - fmaxf(rb.x, cbB.x);
      float ih = fminf(rb.w, cbB.w) - fmaxf(rb.y, cbB.y);
      float inter = fmaxf(iw, 0.0f) * fmaxf(ih, 0.0f);
      float uni = acc1[v] - inter;
      bool sup = (uni > 0.0f) && (inter > NMS_T * uni) && (colB > rowIdx) && (colB < KNC);
      bitsB |= (sup ? 1u : 0u) << v;
    }
  }
  // swap row-halves so ballot lane order == column order for every row
  unsigned bitsA_sw = (unsigned)__shfl_xor((int)bitsA, 16, 32);
  unsigned bitsB_sw = (unsigned)__shfl_xor((int)bitsB, 16, 32);
  #pragma unroll
  for (int r = 0; r < 16; ++r) {
    int v = r & 7;
    unsigned src = (r < 8) ? ((half == 0) ? bitsA : bitsB_sw)
                           : ((half == 0) ? bitsA_sw : bitsB);
    unsigned word = (unsigned)__ballot((src >> v) & 1u);
    if (lane == r) mrow[(size_t)(rowBase + r) * KMS + ct] = word;
  }
}

// ---------------- K5: serial greedy scan (1 wave/image) + top-100 select -------------
// keep mask (188 words) lives in registers, 6 words/lane. Mask rows are
// double-buffered into LDS with global_load_async_to_lds_b128 / s_wait_asynccnt.
__global__ void k_scan_select(char* __restrict__ ws, float* __restrict__ out) {
  extern __shared__ char smem5[];          // [0,2048) 2 row buffers, then kw/sel/cnt
  int b = blockIdx.x, lane = threadIdx.x;
  unsigned ldsbase = (unsigned)(unsigned long long)(void*)smem5;  // LDS byte offset

  // initial keep = valid flags of sorted candidates (cols >= 6000 stay 0)
  const int* sval = (const int*)(ws + O_SVALID) + (size_t)b * KNS;
  unsigned k0 = 0, k1 = 0, k2 = 0, k3 = 0, k4 = 0, k5 = 0;
  for (int w = 0; w < KMW; ++w) {
    int idx = w * 32 + lane;
    bool v = (idx < KNC) && (sval[idx] != 0);
    unsigned m = (unsigned)__ballot(v);
    if ((w & 31) == lane) {
      switch (w >> 5) {
        case 0: k0 = m; break; case 1: k1 = m; break; case 2: k2 = m; break;
        case 3: k3 = m; break; case 4: k4 = m; break; default: k5 = m; break;
      }
    }
  }

  const char* maskB = ws + O_MASK + (size_t)b * KNC * KMS * 4;
  {  // pre-issue row 0 into buffer 0 (two b128 async loads cover words 0..191)
    const char* g = maskB;
    unsigned l0 = ldsbase + lane * 16u;
    asm volatile("global_load_async_to_lds_b128 %0, %1, off" :: "v"(l0), "v"(g + lane * 16) : "memory");
    asm volatile("global_load_async_to_lds_b128 %0, %1, off" :: "v"(l0 + 512u), "v"(g + 512 + lane * 16) : "memory");
  }
  for (int i = 0; i < KNC; ++i) {
    if (i + 1 < KNC) {  // prefetch next row, then wait so row i is resident
      const char* g = maskB + (size_t)(i + 1) * KMS * 4;
      unsigned l0 = ldsbase + ((unsigned)((i + 1) & 1)) * 1024u + lane * 16u;
      asm volatile("global_load_async_to_lds_b128 %0, %1, off" :: "v"(l0), "v"(g + lane * 16) : "memory");
      asm volatile("global_load_async_to_lds_b128 %0, %1, off" :: "v"(l0 + 512u), "v"(g + 512 + lane * 16) : "memory");
      asm volatile("s_wait_asynccnt 0x2" ::: "memory");
      if (i + 2 < KNC) __builtin_prefetch(maskB + (size_t)(i + 2) * KMS * 4, 0, 0);
    } else {
      asm volatile("s_wait_asynccnt 0x0" ::: "memory");
    }
    int w = i >> 5, slot = w >> 5, owner = w & 31;
    unsigned kv;
    switch (slot) {
      case 0: kv = k0; break; case 1: kv = k1; break; case 2: kv = k2; break;
      case 3: kv = k3; break; case 4: kv = k4; break; default: kv = k5; break;
    }
    kv = (unsigned)__shfl((int)kv, owner, 32);
    if ((kv >> (i & 31)) & 1u) {           // row i survives -> suppress its mask
      const unsigned* bp = (const unsigned*)smem5 + (size_t)(i & 1) * 256;
      k0 &= ~bp[lane];       k1 &= ~bp[32 + lane];  k2 &= ~bp[64 + lane];
      k3 &= ~bp[96 + lane];  k4 &= ~bp[128 + lane]; k5 &= ~bp[160 + lane];
    }
  }
  asm volatile("s_wait_asynccnt 0x0" ::: "memory");

  // publish keep words, lane0 picks first 100 kept (already in score order)
  unsigned* kw = (unsigned*)(smem5 + 2048);
  kw[lane] = k0; kw[32 + lane] = k1; kw[64 + lane] = k2;
  kw[96 + lane] = k3; kw[128 + lane] = k4; kw[160 + lane] = k5;
  int* sel = (int*)(smem5 + 2048 + 192 * 4);
  int* pcnt = sel + KTOPK;
  __syncthreads();
  if (lane == 0) {
    int cnt = 0;
    for (int w = 0; w < KMW && cnt < KTOPK; ++w) {
      unsigned m = kw[w];
      while (m && cnt < KTOPK) { int bit = __ffs(m) - 1; sel[cnt++] = w * 32 + bit; m &= m - 1; }
    }
    *pcnt = cnt;
  }
  __syncthreads();
  int cnt = *pcnt;

  const float4* sbox = (const float4*)(ws + O_SBOX) + (size_t)b * KNS;
  const float*  ssc  = (const float*)(ws + O_SSCORE) + (size_t)b * KNS;
  const int*    slb  = (const int*)(ws + O_SLABEL) + (size_t)b * KNS;
  const float*  sln  = (const float*)(ws + O_SLEN) + (size_t)b * KNS;
  float* obox   = out;                                  // [B,100,4]
  float* oscore = out + KB * KTOPK * 4;                 // [B,100]
  int*   olabel = (int*)(out + KB * KTOPK * 5);         // [B,100] int32
  float* olen   = out + KB * KTOPK * 6;                 // [B,100]
  int*   ovalid = (int*)(out + KB * KTOPK * 7);         // [B,100] bool as i32
  for (int k = lane; k < KTOPK; k += 32) {
    int o = b * KTOPK + k;
    if (k < cnt) {
      int p = sel[k];
      float4 bx = sbox[p];
      obox[o * 4 + 0] = bx.x; obox[o * 4 + 1] = bx.y;
      obox[o * 4 + 2] = bx.z; obox[o * 4 + 3] = bx.w;
      oscore[o] = ssc[p]; olabel[o] = slb[p]; olen[o] = sln[p]; ovalid[o] = 1;
    } else {
      obox[o * 4 + 0] = 0; obox[o * 4 + 1] = 0; obox[o * 4 + 2] = 0; obox[o * 4 + 3] = 0;
      oscore[o] = 0; olabel[o] = -1; olen[o] = 0; ovalid[o] = 0;
    }
  }
}

// ---------------- host launcher ----------------
extern "C" void kernel_launch(void* const* d_in, const int* in_sizes, int n_in,
                              void* d_out, int out_size, void* d_ws, size_t ws_size,
                              hipStream_t stream) {
  const float* cls  = (const float*)d_in[0];   // [4,2000,4]
  const float* reg  = (const float*)d_in[1];   // [4,2000,16]
  const float* lenl = (const float*)d_in[2];   // [4,2000,1]
  const float* prop = (const float*)d_in[3];   // [4,2000,4]
  char* ws = (char*)d_ws;

  k_init<<<1, 32, 0, stream>>>(ws);
  k_decode<<<(KB * KP + 255) / 256, 256, 0, stream>>>(cls, reg, lenl, prop, ws);
  k_sort<<<KB, 1024, KNSORT * sizeof(unsigned long long), stream>>>(ws);
  k_gather<<<(KB * KNS + 255) / 256, 256, 0, stream>>>(ws);
  int tiles = KB * (KNC / 16) * KMW;
  k_mask<<<(tiles + 7) / 8, 256, 0, stream>>>(ws);
  k_scan_select<<<KB, 32, 4096, stream>>>(ws, (float*)d_out);
}